// AstroMiNN_55997783605483
// MI455X (gfx1250) — compile-verified
//
#include <hip/hip_runtime.h>
#include <math.h>

#define BATCH 16384
#define FDIM  576
#define HDIM  512
#define NEXP  16
#define ODIM  5
#define KC    32

typedef float v2f __attribute__((ext_vector_type(2)));
typedef float v8f __attribute__((ext_vector_type(8)));

__device__ __forceinline__ float gelu_exact(float x) {
  return 0.5f * x * (1.0f + erff(x * 0.7071067811865475f));
}
__device__ __forceinline__ float sigmoid_(float x) {
  return 1.0f / (1.0f + expf(-x));
}
__device__ __forceinline__ float wave_sum(float v) {
  for (int m = 16; m >= 1; m >>= 1) v += __shfl_xor(v, m, 32);
  return v;
}
__device__ __forceinline__ void wait_async6() {
#if __has_builtin(__builtin_amdgcn_s_wait_asynccnt)
  __builtin_amdgcn_s_wait_asynccnt(6);
#else
  asm volatile("s_wait_asynccnt 6" ::: "memory");
#endif
}
__device__ __forceinline__ void wait_async0() {
#if __has_builtin(__builtin_amdgcn_s_wait_asynccnt)
  __builtin_amdgcn_s_wait_asynccnt(0);
#else
  asm volatile("s_wait_asynccnt 0" ::: "memory");
#endif
}
// per-lane async copy: 16 contiguous global bytes -> 16 contiguous LDS bytes
__device__ __forceinline__ void async_b128(unsigned lds_off, int goff, const void* sbase) {
  asm volatile("global_load_async_to_lds_b128 %0, %1, %2"
               :: "v"(lds_off), "v"(goff), "s"(sbase) : "memory");
}

// ---------------- init: zero output + expert counters ----------------
__global__ __launch_bounds__(256) void k_init(float* out, int* cnt) {
  int i = blockIdx.x * 256 + threadIdx.x;
  if (i < BATCH * ODIM) out[i] = 0.0f;
  if (i < NEXP) cnt[i] = 0;
}

// ---------------- towers: wave-per-row small MLPs ----------------
struct TowerP {
  const float *W1, *b1, *Wm, *bm, *Wa, *ba, *Ws, *bs, *lgm, *lbm, *lga, *lba;
  int din, dh, off;
  int cols[19];
};
struct TowersArg { TowerP t[8]; };

__global__ __launch_bounds__(256) void k_towers(const float* __restrict__ md,
                                                float* __restrict__ af,
                                                TowersArg P) {
  __shared__ float hsh[8][192];
  const int wave = threadIdx.x >> 5, lane = threadIdx.x & 31;
  const int row = blockIdx.x * 8 + wave;
  for (int t = 0; t < 8; ++t) {
    const TowerP& p = P.t[t];
    float x[19];
    for (int i = 0; i < p.din; ++i) x[i] = md[row * 24 + p.cols[i]];
    const int nk = p.dh >> 5;
    float s = 0.f, s2 = 0.f;
    for (int ii = 0; ii < nk; ++ii) {
      int k = lane + (ii << 5);
      float a = p.b1[k];
      for (int i = 0; i < p.din; ++i) a += x[i] * p.W1[i * p.dh + k];
      a = gelu_exact(a);
      hsh[wave][k] = a;
      s += a; s2 += a * a;
    }
    s = wave_sum(s); s2 = wave_sum(s2);
    const float invd = 1.0f / (float)p.dh;
    const float mean = s * invd;
    const float rstd = rsqrtf(s2 * invd - mean * mean + 1e-5f);
    __syncthreads();
    for (int rep = 0; rep < 2; ++rep) {
      int j = lane + (rep << 5);
      float am = 0.f, ag = 0.f;
      for (int k = 0; k < p.dh; ++k) {
        float c = (hsh[wave][k] - mean) * rstd;
        am += (c * p.lgm[k] + p.lbm[k]) * p.Wm[k * 64 + j];
        ag += (c * p.lga[k] + p.lba[k]) * p.Wa[k * 64 + j];
      }
      float skip = p.bs[j];
      for (int i = 0; i < p.din; ++i) skip += x[i] * p.Ws[i * 64 + j];
      af[(size_t)row * FDIM + p.off + j] =
          (am + p.bm[j]) * sigmoid_(ag + p.ba[j]) + skip;
    }
    __syncthreads();
  }
  // image_feats block (cols 448..511) is zeros
  af[(size_t)row * FDIM + 448 + lane] = 0.0f;
  af[(size_t)row * FDIM + 480 + lane] = 0.0f;
}

// ---------------- router + top-2 dispatch ----------------
__global__ __launch_bounds__(256) void k_router(const float* __restrict__ af,
    const float* __restrict__ Wr1, const float* __restrict__ br1,
    const float* __restrict__ Wr2, const float* __restrict__ br2,
    int* __restrict__ cnt, int* __restrict__ ridx, float* __restrict__ rw) {
  __shared__ float afs[8][FDIM];
  __shared__ float ts[8][288];
  __shared__ float fw[8][16];
  const int wave = threadIdx.x >> 5, lane = threadIdx.x & 31;
  const int row = blockIdx.x * 8 + wave;
  for (int i = lane; i < FDIM; i += 32) afs[wave][i] = af[(size_t)row * FDIM + i];
  __syncthreads();
  for (int r = 0; r < 9; ++r) {                 // 288 = 9*32
    int j = lane + (r << 5);
    float a = br1[j];
    for (int k = 0; k < FDIM; ++k) a += afs[wave][k] * Wr1[k * 288 + j];
    ts[wave][j] = tanhf(a);
  }
  __syncthreads();
  if (lane < 16) {
    float a = br2[lane];
    for (int k = 0; k < 288; ++k) a += ts[wave][k] * Wr2[k * 16 + lane];
    fw[wave][lane] = sigmoid_(a);
  }
  __syncthreads();
  if (lane == 0) {
    int i1 = 0; float w1 = fw[wave][0];
    for (int l = 1; l < 16; ++l)
      if (fw[wave][l] > w1) { w1 = fw[wave][l]; i1 = l; }
    int i2 = (i1 == 0) ? 1 : 0; float w2 = fw[wave][i2];
    for (int l = 0; l < 16; ++l)
      if (l != i1 && fw[wave][l] > w2) { w2 = fw[wave][l]; i2 = l; }
    int s1 = atomicAdd(&cnt[i1], 1);
    ridx[i1 * BATCH + s1] = row; rw[i1 * BATCH + s1] = w1;
    int s2 = atomicAdd(&cnt[i2], 1);
    ridx[i2 * BATCH + s2] = row; rw[i2 * BATCH + s2] = w2;
  }
}

// ---------------- expert GEMM: gathered rows, WMMA f32 16x16x4 ----------------
// block tile: 64 (slots) x 128 (cols), K chunked by 32, double-buffered LDS
// filled by GLOBAL_LOAD_ASYNC_TO_LDS_B128 (ASYNCcnt pipeline).
// 8 waves in 2x4 grid, each owns a 32x32 tile = 2x2 WMMA accumulators.
__global__ __launch_bounds__(256) void k_expert_gemm(
    const float* __restrict__ af, const int* __restrict__ ridx,
    const int* __restrict__ cnt, const float* __restrict__ W1all,
    const float* __restrict__ b1all, float* __restrict__ h, int e) {
  __shared__ __attribute__((aligned(16))) float As[2][64][36]; // row=144B, 16B aligned
  __shared__ __attribute__((aligned(16))) float Bs[2][32][128];
  __shared__ int rix[64];
  const int count = cnt[e];
  const int m0 = blockIdx.y * 64;
  if (m0 >= count) return;
  const int n0 = blockIdx.x * 128;
  const float* __restrict__ W  = W1all + (size_t)e * FDIM * HDIM;
  const float* __restrict__ b1 = b1all + e * HDIM;
  const int*   __restrict__ rl = ridx + e * BATCH;
  const int tid = threadIdx.x;
  const int wave = tid >> 5, lane = tid & 31;
  if (tid < 64) {
    int s = m0 + tid;
    rix[tid] = rl[(s < count) ? s : (count - 1)];  // clamp: dummy rows harmless
  }
  __syncthreads();

  // ---- async-copy address setup (per-thread, 6 x B128 per K-chunk) ----
  const unsigned asBase = (unsigned)(size_t)(void*)&As[0][0][0];
  const unsigned bsBase = (unsigned)(size_t)(void*)&Bs[0][0][0];
  const int q  = tid & 7;               // 16B chunk within an A row (8*16B=128B)
  const int r0 = tid >> 3;              // A rows r0 and r0+32
  const int row0 = rix[r0], row1 = rix[r0 + 32];
  const int aG0 = row0 * (FDIM * 4) + q * 16;    // + k0*4
  const int aG1 = row1 * (FDIM * 4) + q * 16;
  const unsigned aL0 = (unsigned)(r0 * 144 + q * 16);          // + buf*9216
  const unsigned aL1 = (unsigned)((r0 + 32) * 144 + q * 16);
  int      bG[4];
  unsigned bL[4];
  for (int j = 0; j < 4; ++j) {         // chunk c = tid + 256j -> kk = wave+8j
    int kk = wave + 8 * j;
    bG[j] = kk * (HDIM * 4) + n0 * 4 + lane * 16;              // + k0*HDIM*4
    bL[j] = (unsigned)(kk * 512 + lane * 16);                  // + buf*16384
  }
  auto issue_chunk = [&](int k0, int buf) {
    const unsigned aOff = asBase + (buf ? 9216u : 0u);
    const unsigned bOff = bsBase + (buf ? 16384u : 0u);
    const int ak = k0 * 4;
    const int bk = k0 * (HDIM * 4);
    async_b128(aOff + aL0, aG0 + ak, (const void*)af);
    async_b128(aOff + aL1, aG1 + ak, (const void*)af);
    async_b128(bOff + bL[0], bG[0] + bk, (const void*)W);
    async_b128(bOff + bL[1], bG[1] + bk, (const void*)W);
    async_b128(bOff + bL[2], bG[2] + bk, (const void*)W);
    async_b128(bOff + bL[3], bG[3] + bk, (const void*)W);
  };

  const int wm = wave >> 2, wn = wave & 3;
  const int ml = lane & 15;
  const int kb = (lane >> 4) << 1;      // K pair base: 0 or 2
  v8f c[2][2];
  for (int tm = 0; tm < 2; ++tm)
    for (int tn = 0; tn < 2; ++tn)
      for (int v = 0; v < 8; ++v) c[tm][tn][v] = 0.0f;

  issue_chunk(0, 0);
  issue_chunk(KC, 1);
  const int NCHUNK = FDIM / KC;                 // 18
  for (int i = 0; i < NCHUNK; ++i) {
    if (i < NCHUNK - 1) wait_async6();          // chunk i landed, i+1 in flight
    else                wait_async0();
    __syncthreads();
    const int buf = i & 1;
    const float (*A)[36]  = As[buf];
    const float (*B)[128] = Bs[buf];
    for (int kk = 0; kk < KC; kk += 4) {
      v2f a[2], b[2];
      for (int tm = 0; tm < 2; ++tm) {
        int mr = wm * 32 + tm * 16 + ml;
        a[tm].x = A[mr][kk + kb];
        a[tm].y = A[mr][kk + kb + 1];
      }
      for (int tn = 0; tn < 2; ++tn) {
        int nc = wn * 32 + tn * 16 + ml;
        b[tn].x = B[kk + kb][nc];
        b[tn].y = B[kk + kb + 1][nc];
      }
      for (int tm = 0; tm < 2; ++tm)
        for (int tn = 0; tn < 2; ++tn)
          c[tm][tn] = __builtin_amdgcn_wmma_f32_16x16x4_f32(
              false, a[tm], false, b[tn], (short)0, c[tm][tn], false, false);
    }
    __syncthreads();                            // all waves done reading buf
    if (i + 2 < NCHUNK) issue_chunk((i + 2) * KC, buf);
  }
  // epilogue: + bias, exact GELU, store slot-indexed h
  for (int tm = 0; tm < 2; ++tm)
    for (int tn = 0; tn < 2; ++tn)
      for (int v = 0; v < 8; ++v) {
        int slot = m0 + wm * 32 + tm * 16 + v + ((lane >> 4) << 3);
        int col  = n0 + wn * 32 + tn * 16 + ml;
        float xv = c[tm][tn][v] + b1[col];
        h[(size_t)slot * HDIM + col] = gelu_exact(xv);
      }
}

// ---------------- expert tail: LN x2, heads, gate, skip, weighted add ----------------
__global__ __launch_bounds__(256) void k_expert_tail(
    const float* __restrict__ af, const int* __restrict__ ridx,
    const float* __restrict__ rw, const int* __restrict__ cnt,
    const float* __restrict__ h,
    const float* __restrict__ lmg, const float* __restrict__ lmb,
    const float* __restrict__ lag, const float* __restrict__ lab,
    const float* __restrict__ Wm, const float* __restrict__ bm,
    const float* __restrict__ Wa, const float* __restrict__ ba,
    const float* __restrict__ Ws, const float* __restrict__ bs,
    float* __restrict__ out, int e) {
  const int wave = threadIdx.x >> 5, lane = threadIdx.x & 31;
  const int slot = blockIdx.x * 8 + wave;
  if (slot >= cnt[e]) return;
  const int row = ridx[e * BATCH + slot];
  const float wgt = rw[e * BATCH + slot];
  const float* hr  = h + (size_t)slot * HDIM;
  const float* gm  = lmg + e * HDIM; const float* bmn = lmb + e * HDIM;
  const float* ga  = lag + e * HDIM; const float* ban = lab + e * HDIM;
  const float* Wm_ = Wm + (size_t)e * HDIM * ODIM;
  const float* Wa_ = Wa + (size_t)e * HDIM * ODIM;
  const float* Ws_ = Ws + (size_t)e * FDIM * ODIM;
  float hv[16];
  float s = 0.f, s2 = 0.f;
  for (int i = 0; i < 16; ++i) {
    float v = hr[lane + (i << 5)];
    hv[i] = v; s += v; s2 += v * v;
  }
  s = wave_sum(s); s2 = wave_sum(s2);
  const float mean = s * (1.0f / 512.0f);
  const float rstd = rsqrtf(s2 * (1.0f / 512.0f) - mean * mean + 1e-5f);
  float aM[5] = {}, aG[5] = {}, aS[5] = {};
  for (int i = 0; i < 16; ++i) {
    int k = lane + (i << 5);
    float cv = (hv[i] - mean) * rstd;
    float hm = cv * gm[k] + bmn[k];
    float ha = cv * ga[k] + ban[k];
    for (int j = 0; j < 5; ++j) {
      aM[j] += hm * Wm_[k * 5 + j];
      aG[j] += ha * Wa_[k * 5 + j];
    }
  }
  const float* xr = af + (size_t)row * FDIM;
  for (int i = 0; i < 18; ++i) {                 // 576 = 18*32
    int k = lane + (i << 5);
    float xv = xr[k];
    for (int j = 0; j < 5; ++j) aS[j] += xv * Ws_[k * 5 + j];
  }
  for (int j = 0; j < 5; ++j) {
    aM[j] = wave_sum(aM[j]); aG[j] = wave_sum(aG[j]); aS[j] = wave_sum(aS[j]);
  }
  if (lane == 0) {
    for (int j = 0; j < 5; ++j) {
      float mn = aM[j] + bm[e * ODIM + j];
      float gt = sigmoid_(aG[j] + ba[e * ODIM + j]);
      out[row * ODIM + j] += wgt * (mn * gt + aS[j] + bs[e * ODIM + j]);
    }
  }
}

// ---------------- host launch ----------------
extern "C" void kernel_launch(void* const* d_in, const int* in_sizes, int n_in,
                              void* d_out, int out_size, void* d_ws, size_t ws_size,
                              hipStream_t stream) {
  (void)in_sizes; (void)n_in; (void)out_size; (void)ws_size;
  // jax pytree flatten order (dict keys sorted): metadata, experts(12), router(4), towers(8x12)
  const float* md   = (const float*)d_in[0];
  const float* eW1  = (const float*)d_in[1];
  const float* eWa  = (const float*)d_in[2];
  const float* eWm  = (const float*)d_in[3];
  const float* eWs  = (const float*)d_in[4];
  const float* eb1  = (const float*)d_in[5];
  const float* eba  = (const float*)d_in[6];
  const float* ebm  = (const float*)d_in[7];
  const float* ebs  = (const float*)d_in[8];
  const float* elab = (const float*)d_in[9];
  const float* elag = (const float*)d_in[10];
  const float* elmb = (const float*)d_in[11];
  const float* elmg = (const float*)d_in[12];
  const float* Wr1  = (const float*)d_in[13];
  const float* Wr2  = (const float*)d_in[14];
  const float* br1  = (const float*)d_in[15];
  const float* br2  = (const float*)d_in[16];

  char* ws = (char*)d_ws;
  float* af   = (float*)(ws);                                      // 37,748,736 B
  int*   cnt  = (int*)  (ws + 37748736);                           // 256 B
  int*   ridx = (int*)  (ws + 37748736 + 256);                     // 1,048,576 B
  float* rwts = (float*)(ws + 37748736 + 256 + 1048576);           // 1,048,576 B
  float* hbuf = (float*)(ws + 37748736 + 256 + 2097152);           // 33,554,432 B
  float* out  = (float*)d_out;

  TowersArg TA;
  auto fill = [&](int slot, int base, int din, int dh, int off, const int* cols) {
    TowerP& p = TA.t[slot];
    p.W1  = (const float*)d_in[base + 0];
    p.Wa  = (const float*)d_in[base + 1];
    p.Wm  = (const float*)d_in[base + 2];
    p.Ws  = (const float*)d_in[base + 3];
    p.b1  = (const float*)d_in[base + 4];
    p.ba  = (const float*)d_in[base + 5];
    p.bm  = (const float*)d_in[base + 6];
    p.bs  = (const float*)d_in[base + 7];
    p.lba = (const float*)d_in[base + 8];
    p.lga = (const float*)d_in[base + 9];
    p.lbm = (const float*)d_in[base + 10];
    p.lgm = (const float*)d_in[base + 11];
    p.din = din; p.dh = dh; p.off = off;
    for (int i = 0; i < din; ++i) p.cols[i] = cols[i];
    for (int i = din; i < 19; ++i) p.cols[i] = 0;
  };
  static const int c_nst1[]    = {0, 2};
  static const int c_nst2[]    = {1, 3};
  static const int c_spatial[] = {2, 3, 4};
  static const int c_psf[]     = {5, 14};
  static const int c_mag[]     = {6, 9, 10, 13, 15, 17, 18};
  static const int c_coord[]   = {7, 8};
  static const int c_mega[]    = {0,1,2,3,4,5,6,7,8,9,10,11,12,13,14,15,16,17,18};
  static const int c_lc[]      = {6, 9, 10, 13, 15, 17, 18, 19, 20, 21, 22, 23};
  // tower bases: coord=17, lc=29, mag=41, mega=53, nst1=65, nst2=77, psf=89, spatial=101
  fill(0, 65,  2,  64,   0, c_nst1);
  fill(1, 77,  2,  64,  64, c_nst2);
  fill(2, 101, 3,  64, 128, c_spatial);
  fill(3, 89,  2,  64, 192, c_psf);
  fill(4, 41,  7, 128, 256, c_mag);
  fill(5, 17,  2,  64, 320, c_coord);
  fill(6, 53, 19, 128, 384, c_mega);
  fill(7, 29, 12, 192, 512, c_lc);

  hipLaunchKernelGGL(k_init, dim3(320), dim3(256), 0, stream, out, cnt);
  hipLaunchKernelGGL(k_towers, dim3(BATCH / 8), dim3(256), 0, stream, md, af, TA);
  hipLaunchKernelGGL(k_router, dim3(BATCH / 8), dim3(256), 0, stream,
                     af, Wr1, br1, Wr2, br2, cnt, ridx, rwts);
  for (int e = 0; e < NEXP; ++e) {
    hipLaunchKernelGGL(k_expert_gemm, dim3(4, BATCH / 64), dim3(256), 0, stream,
                       af, ridx, cnt, eW1, eb1, hbuf, e);
    hipLaunchKernelGGL(k_expert_tail, dim3(BATCH / 8), dim3(256), 0, stream,
                       af, ridx, rwts, cnt, hbuf,
                       elmg, elmb, elag, elab, eWm, ebm, eWa, eba, eWs, ebs,
                       out, e);
  }
}